// Noise_Regressor_9637906612688
// MI455X (gfx1250) — compile-verified
//
#include <hip/hip_runtime.h>
#include <math.h>

#define SEQ_LEN   4096
#define D_MODEL   1024
#define NUM_AXES  72
#define NPAR      (12 * NUM_AXES)   // 864
#define T_PROP    600
#define HIST      (T_PROP - 1)      // 599
#define JCHUNK    256
#define WIN       (JCHUNK + HIST)   // 855
#define LN_EPS    1e-5f

typedef float v2f __attribute__((ext_vector_type(2)));
typedef float v8f __attribute__((ext_vector_type(8)));

__device__ __forceinline__ float softplus_f(float x) {
    // numerically stable log1p(exp(x))
    return fmaxf(x, 0.0f) + log1pf(__expf(-fabsf(x)));
}

// ---------------------------------------------------------------------------
// Kernel 1: LayerNorm, one block (256 threads) per token, two-pass in regs.
// ---------------------------------------------------------------------------
__global__ void __launch_bounds__(256)
ln_kernel(const float* __restrict__ x, const float* __restrict__ gamma,
          const float* __restrict__ beta, float* __restrict__ h) {
    __shared__ float red[256];
    const int s   = blockIdx.x;
    const int tid = threadIdx.x;
    const float* xr = x + (size_t)s * D_MODEL;

    float v[4];
    float sum = 0.f;
#pragma unroll
    for (int i = 0; i < 4; ++i) { v[i] = xr[tid + 256 * i]; sum += v[i]; }
    red[tid] = sum; __syncthreads();
    for (int off = 128; off > 0; off >>= 1) {
        if (tid < off) red[tid] += red[tid + off];
        __syncthreads();
    }
    const float mu = red[0] * (1.0f / D_MODEL);
    __syncthreads();

    float s2 = 0.f;
#pragma unroll
    for (int i = 0; i < 4; ++i) { float d = v[i] - mu; s2 += d * d; }
    red[tid] = s2; __syncthreads();
    for (int off = 128; off > 0; off >>= 1) {
        if (tid < off) red[tid] += red[tid + off];
        __syncthreads();
    }
    const float var  = red[0] * (1.0f / D_MODEL);
    const float rstd = rsqrtf(var + LN_EPS);

    float* hr = h + (size_t)s * D_MODEL;
#pragma unroll
    for (int i = 0; i < 4; ++i) {
        int idx = tid + 256 * i;
        hr[idx] = (v[i] - mu) * rstd * gamma[idx] + beta[idx];
    }
}

// ---------------------------------------------------------------------------
// Kernel 2: p = h @ W^T + b via full-precision V_WMMA_F32_16X16X4_F32.
// Each wave owns one 16(M) x 32(N) output tile: two 16x16 WMMA accumulators,
// A operand reused across both. 6912 waves total, 4 waves/block.
// A layout (ISA 7.12.2, 32-bit A 16x4): lane L holds A[M=L%16][k0 + 2*(L>>4) + v]
// B layout mirrors: lane L holds B[k0 + 2*(L>>4) + v][N=L%16]; B[k][n] = W[n][k].
// ---------------------------------------------------------------------------
__global__ void __launch_bounds__(128)
gemm_wmma_kernel(const float* __restrict__ h, const float* __restrict__ W,
                 const float* __restrict__ bias, float* __restrict__ p) {
    const int wave  = ((int)blockIdx.x * 4) + ((int)threadIdx.x >> 5);
    const int lane  = threadIdx.x & 31;
    const int mTile = wave & 255;   // 256 M-tiles
    const int nPair = wave >> 8;    // 27 N-pairs (864 / 32)

    const int mrow  = mTile * 16 + (lane & 15);
    const int khalf = (lane >> 4) * 2;
    const int n0    = nPair * 32 + (lane & 15);
    const int n1    = n0 + 16;

    const float* arow  = h + (size_t)mrow * D_MODEL + khalf;
    const float* brow0 = W + (size_t)n0   * D_MODEL + khalf;
    const float* brow1 = W + (size_t)n1   * D_MODEL + khalf;

    v8f acc0 = {}; v8f acc1 = {};
    for (int k = 0; k < D_MODEL; k += 4) {
        v2f a  = *(const v2f*)(arow  + k);
        v2f b0 = *(const v2f*)(brow0 + k);
        v2f b1 = *(const v2f*)(brow1 + k);
        acc0 = __builtin_amdgcn_wmma_f32_16x16x4_f32(false, a, false, b0,
                                                     (short)0, acc0, false, false);
        acc1 = __builtin_amdgcn_wmma_f32_16x16x4_f32(false, a, false, b1,
                                                     (short)0, acc1, false, false);
    }

    // D layout: VGPR v -> lanes 0-15: (M=v, N=lane); lanes 16-31: (M=v+8, N=lane-16)
    const int mbase = mTile * 16 + 8 * (lane >> 4);
    const int nc    = nPair * 32 + (lane & 15);
    const float bi0 = bias[nc];
    const float bi1 = bias[nc + 16];
#pragma unroll
    for (int d = 0; d < 8; ++d) {
        const int mm = mbase + d;
        p[(size_t)mm * NPAR + nc]      = acc0[d] + bi0;
        p[(size_t)mm * NPAR + nc + 16] = acc1[d] + bi1;
    }
}

// ---------------------------------------------------------------------------
// Kernel 3: derive oscillator params once per (s, axis) and emit outputs 2..5.
// omega1 = sqrt(4k - d^2)/2 with k = d^2/4 + sp(p0)  ==  sqrt(sp(p0)).
// Derived params packed as TWO float4 per (axis, s):
//   der[(axis*S + s)*8 + {0..3}] = {c,   d/2,   omega1,   phi  }   (linear osc)
//   der[(axis*S + s)*8 + {4..7}] = {c_th,d_th/2,omega1_th,phi_th}  (angular osc)
// Thread index is axis-major so der stores and output-plane stores coalesce.
// ---------------------------------------------------------------------------
__global__ void __launch_bounds__(256)
params_kernel(const float* __restrict__ p, float* __restrict__ der,
              float* __restrict__ out) {
    const int idx = blockIdx.x * blockDim.x + threadIdx.x;
    if (idx >= SEQ_LEN * NUM_AXES) return;
    const int axis = idx / SEQ_LEN;       // axis-major
    const int s    = idx - axis * SEQ_LEN;

    const float* pr = p + (size_t)s * NPAR + axis;   // stride 72 between params
    const float p0  = pr[0 * 72], p1  = pr[1 * 72], p2  = pr[2 * 72];
    const float p3  = pr[3 * 72], p4  = pr[4 * 72], p5  = pr[5 * 72];
    const float p6  = pr[6 * 72], p7  = pr[7 * 72], p8  = pr[8 * 72];
    const float p9  = pr[9 * 72], p10 = pr[10 * 72], p11 = pr[11 * 72];

    const float d_lin  = softplus_f(p1);
    const float om_lin = sqrtf(softplus_f(p0));
    const float d_th   = softplus_f(p3);
    const float om_th  = sqrtf(softplus_f(p2));

    const size_t PL = (size_t)SEQ_LEN * NUM_AXES;
    const size_t o  = (size_t)idx;                    // == axis*SEQ_LEN + s
    float4* dst = (float4*)(der + o * 8);
    dst[0] = make_float4(p4, 0.5f * d_lin, om_lin, p6);
    dst[1] = make_float4(p5, 0.5f * d_th,  om_th,  p7);

    out[1 * PL + o] = p8;               // acc_base
    out[2 * PL + o] = softplus_f(p9);   // acc_std
    out[3 * PL + o] = p10;              // gyro_base
    out[4 * PL + o] = softplus_f(p11);  // gyro_std
}

// ---------------------------------------------------------------------------
// Kernel 4: gather-form segment sum (race-free, no atomics).
// out[axis][j] = sum_{t=0..min(599,j)} c[j-t]*exp(-d2[j-t]*t)*sin(om[j-t]*t+phi[j-t])
//              + (theta oscillator).
// Per block: one axis, 256 consecutive j; stage 855 positions x 2 float4 in LDS.
// Inner loop: 2x ds_load_b128 + 2x v_exp_f32 + 2x v_sin_f32 per term.
// ---------------------------------------------------------------------------
__global__ void __launch_bounds__(256)
accum_kernel(const float* __restrict__ der, float* __restrict__ out) {
    __shared__ float4 ldsL[WIN];   // linear-osc params   (855 * 16 B)
    __shared__ float4 ldsA[WIN];   // angular-osc params  (855 * 16 B)

    const int axis   = blockIdx.x >> 4;   // 16 chunks per axis
    const int chunk  = blockIdx.x & 15;
    const int tid    = threadIdx.x;
    const int jstart = chunk * JCHUNK;
    const float4* base = (const float4*)(der + (size_t)axis * SEQ_LEN * 8);

    const float4 zero4 = make_float4(0.f, 0.f, 0.f, 0.f);
    for (int i = tid; i < WIN; i += 256) {
        const int src = jstart - HIST + i;
        if (src >= 0) {
            ldsL[i] = base[2 * src + 0];
            ldsA[i] = base[2 * src + 1];
        } else {
            ldsL[i] = zero4;
            ldsA[i] = zero4;
        }
    }
    __syncthreads();

    const int j    = jstart + tid;
    const int tmax = (j < HIST) ? j : HIST;
    float acc = 0.0f;
    for (int t = 0; t <= tmax; ++t) {
        const int   i  = tid + HIST - t;      // index of source token j - t
        const float tf = (float)t;
        const float4 L = ldsL[i];             // {c, d/2, omega1, phi}
        const float4 A = ldsA[i];             // {c_th, d_th/2, omega1_th, phi_th}
        const float e1 = __expf(-L.y * tf);
        const float e2 = __expf(-A.y * tf);
        const float s1 = __sinf(fmaf(L.z, tf, L.w));
        const float s2 = __sinf(fmaf(A.z, tf, A.w));
        acc = fmaf(L.x * e1, s1, acc);
        acc = fmaf(A.x * e2, s2, acc);
    }
    out[(size_t)axis * SEQ_LEN + j] = acc;
}

// ---------------------------------------------------------------------------
extern "C" void kernel_launch(void* const* d_in, const int* in_sizes, int n_in,
                              void* d_out, int out_size, void* d_ws, size_t ws_size,
                              hipStream_t stream) {
    const float* x     = (const float*)d_in[0];   // hidden_states (1,4096,1024)
    const float* gamma = (const float*)d_in[1];   // ln_gamma (1024)
    const float* beta  = (const float*)d_in[2];   // ln_beta  (1024)
    const float* W     = (const float*)d_in[3];   // W (864,1024)
    const float* bvec  = (const float*)d_in[4];   // b (864)
    float* out = (float*)d_out;

    float* h   = (float*)d_ws;                          // 4096*1024 floats
    float* p   = h + (size_t)SEQ_LEN * D_MODEL;         // 4096*864 floats
    float* der = p + (size_t)SEQ_LEN * NPAR;            // 8 * 72*4096 floats

    ln_kernel<<<SEQ_LEN, 256, 0, stream>>>(x, gamma, beta, h);

    // 256 M-tiles * 27 N-pairs = 6912 waves, 4 waves (128 thr) per block
    gemm_wmma_kernel<<<6912 / 4, 128, 0, stream>>>(h, W, bvec, p);

    const int np = SEQ_LEN * NUM_AXES;
    params_kernel<<<(np + 255) / 256, 256, 0, stream>>>(p, der, out);

    accum_kernel<<<NUM_AXES * 16, 256, 0, stream>>>(der, out);
}